// mem_update_66477503807495
// MI455X (gfx1250) — compile-verified
//
#include <hip/hip_runtime.h>

// Leaky integrate-and-fire forward scan, MI455X (gfx1250, wave32).
//
// x:   [T=8, B*C*H*W = 4,194,304] fp32   (flat per-timestep plane)
// out: [T=8, 4,194,304] fp32 spikes in {0,1}
//
// Pure HBM-streaming kernel: 256 MiB total traffic / 23.3 TB/s ~= 11.5 us
// floor; ~0.17 GFLOP of VALU work is negligible, so no matrix-op path applies.
// Each thread owns 4 consecutive spatial elements (float4) for ALL 8
// timesteps: the scan state (mem, spike) lives in VGPRs and every HBM byte is
// touched exactly once. All 8 B128 loads are issued before the recurrence
// (addresses are state-independent) for maximal memory-level parallelism.
// Loads/stores use the gfx1250 non-temporal policy (TH=NT) since nothing is
// reused and the stream exceeds the 192 MB L2.
//
// Addressing refinement: per-timestep plane bases are hoisted as UNIFORM
// pointers so the compiler emits saddr(SGPR-pair) + shared 32-bit voffset
// scale_offset forms for all 16 vmem ops (SALU plane math instead of per-plane
// v_add_nc_u64 chains) -> lower VGPR pressure, more resident waves.

#define LIF_THRESH 1.0f
#define LIF_DECAY  0.25f
#define LIF_T      8

typedef float f32x4 __attribute__((ext_vector_type(4)));

__global__ __launch_bounds__(256) void lif_scan_kernel(
    const f32x4* __restrict__ x,   // [T * n4] float4
    f32x4* __restrict__ out,       // [T * n4] float4
    int n4)                        // float4 elements per timestep plane
{
    const int i = blockIdx.x * blockDim.x + threadIdx.x;
    if (i >= n4) return;

    // Issue all T independent 128-bit NT loads up front. Each plane base
    // (x + t*n4) is wave-uniform -> SGPR-pair saddr; i is the shared voffset.
    f32x4 xv[LIF_T];
#pragma unroll
    for (int t = 0; t < LIF_T; ++t) {
        const f32x4* __restrict__ xt = x + (size_t)t * (size_t)n4;  // uniform
        xv[t] = __builtin_nontemporal_load(xt + i);
    }

    f32x4 mem   = {0.0f, 0.0f, 0.0f, 0.0f};
    f32x4 spike = {0.0f, 0.0f, 0.0f, 0.0f};

#pragma unroll
    for (int t = 0; t < LIF_T; ++t) {
#pragma unroll
        for (int c = 0; c < 4; ++c) {
            // mem = mem*DECAY*(1 - spike) + x   (spike is exactly 0 or 1, so
            // this matches the reference bit-for-bit: term is 0 or mem*0.25)
            float m = mem[c] * (LIF_DECAY * (1.0f - spike[c])) + xv[t][c];
            float s = (m >= LIF_THRESH) ? 1.0f : 0.0f;   // v_cmp + v_cndmask
            mem[c]   = m;
            spike[c] = s;
        }
        f32x4* __restrict__ ot = out + (size_t)t * (size_t)n4;      // uniform
        __builtin_nontemporal_store(spike, ot + i);
    }
}

extern "C" void kernel_launch(void* const* d_in, const int* in_sizes, int n_in,
                              void* d_out, int out_size, void* d_ws, size_t ws_size,
                              hipStream_t stream) {
    (void)n_in; (void)out_size; (void)d_ws; (void)ws_size;

    const f32x4* x   = (const f32x4*)d_in[0];
    f32x4*       out = (f32x4*)d_out;

    // in_sizes[0] = T * B * C * H * W = 8 * 4,194,304 = 33,554,432 floats.
    const int total = in_sizes[0];
    const int n4    = (total / LIF_T) / 4;   // float4 per timestep plane = 1,048,576

    const int threads = 256;                 // 8 wave32 per block
    const int blocks  = (n4 + threads - 1) / threads;   // 4096 blocks

    lif_scan_kernel<<<blocks, threads, 0, stream>>>(x, out, n4);
}